// LNCCLoss3D_16088947491549
// MI455X (gfx1250) — compile-verified
//
#include <hip/hip_runtime.h>
#include <hip/hip_bf16.h>
#include <stdint.h>

// ---------------- problem geometry ----------------
#define D_ 160
#define H_ 192
#define W_ 160
#define B_ 2
#define TD 16
#define THh 16
#define TW 16
#define HD (TD + 8)              // 24 (halo depth)
#define HHy (THh + 8)            // 24 (halo height)
#define HW (TW + 8)              // 24 (halo width)
#define HALO_N (HD * HHy * HW)   // 13824 elements per channel
#define ROWS (HD * HHy)          // 576 (dz,hy) rows after W-reduction
#define NVOXF 9830400.0f         // 2*160*192*160

// LDS layout (floats): xs[13824] ys[13824] b1[576*16] b2[24*16*16] red[256]
#define LDS_FLOATS (2 * HALO_N + ROWS * TW + HD * THh * TW + 256)

typedef float v2f __attribute__((ext_vector_type(2)));
typedef float v8f __attribute__((ext_vector_type(8)));
typedef unsigned int u32x4 __attribute__((ext_vector_type(4)));
typedef int i32x8 __attribute__((ext_vector_type(8)));
typedef int i32x4 __attribute__((ext_vector_type(4)));

// Build TDM descriptor group0 for a 3D tile load (count=1, type=2("image")).
__device__ inline u32x4 tdm_group0(uint32_t lds_addr, uint64_t gaddr) {
  u32x4 g;
  g.x = 1u;                                            // count=1, user mode
  g.y = lds_addr;                                      // LDS byte offset
  g.z = (uint32_t)(gaddr & 0xFFFFFFFFull);             // global_addr[31:0]
  g.w = (uint32_t)((gaddr >> 32) & 0x01FFFFFFull)      // global_addr[56:32]
        | 0x80000000u;                                 // type=2 at bits 127:126
  return g;
}

// group1: data_size=4B, tensor dims = remaining extent from tile start
// (upper OOB reads return zero -> free zero padding), tile 24x24x24,
// strides: row=W_, plane=H_*W_.
__device__ inline i32x8 tdm_group1(uint32_t td0, uint32_t td1) {
  const uint32_t s1 = (uint32_t)(H_ * W_);             // 30720
  i32x8 g;
  g[0] = (int)(2u << 16);                              // data_size = 4 bytes
  g[1] = (int)((td0 & 0xFFFFu) << 16);                 // tensor_dim0 lo
  g[2] = (int)((td0 >> 16) | ((td1 & 0xFFFFu) << 16)); // td0 hi | td1 lo
  g[3] = (int)((td1 >> 16) | ((uint32_t)HW << 16));    // td1 hi | tile_dim0=24
  g[4] = (int)((uint32_t)HHy | ((uint32_t)HD << 16));  // tile_dim1 | tile_dim2
  g[5] = (int)(uint32_t)W_;                            // tensor_dim0_stride lo
  g[6] = (int)((s1 & 0xFFFFu) << 16);                  // stride0 hi=0 | stride1 lo
  g[7] = (int)(s1 >> 16);                              // stride1 hi
  return g;
}

__global__ void __launch_bounds__(256) lncc_main(const float* __restrict__ x,
                                                 const float* __restrict__ y,
                                                 float* __restrict__ ws) {
  extern __shared__ float smem[];
  float* xs  = smem;
  float* ys  = smem + HALO_N;
  float* b1  = smem + 2 * HALO_N;          // [ROWS][16]  W-summed channel
  float* b2  = b1 + ROWS * TW;             // [24][16][16] W+H-summed channel
  float* red = b2 + HD * THh * TW;         // [256] reduction scratch

  const int tid  = threadIdx.x;
  const int wave = tid >> 5;
  const int lane = tid & 31;
  const int half = lane >> 4;
  const int nn   = lane & 15;

  const int wt = blockIdx.x;               // 0..9
  const int ht = blockIdx.y;               // 0..11
  const int zt = blockIdx.z;               // 0..19  (b * Dtiles + dt)
  const int b  = zt / (D_ / TD);
  const int dt = zt % (D_ / TD);
  const int d0 = dt * TD, h0 = ht * THh, w0 = wt * TW;

  // -------- Stage A: halo tile load (Global -> LDS) --------
  const bool tdm_ok = (d0 > 0) && (h0 > 0) && (w0 > 0);  // no lower clip needed
  if (tdm_ok) {
    if (tid < 32) {                        // wave 0 issues both DMAs
      const int sd = d0 - 4, sh = h0 - 4, sw = w0 - 4;
      const size_t eoff = (((size_t)b * D_ + sd) * H_ + sh) * (size_t)W_ + sw;
      const uint32_t td0 = (uint32_t)(W_ - sw);
      const uint32_t td1 = (uint32_t)(H_ - sh);
      const uint32_t td2 = (uint32_t)(D_ - sd);
      i32x8 g1 = tdm_group1(td0, td1);
      i32x4 g2; g2[0] = (int)td2; g2[1] = 0; g2[2] = 0; g2[3] = 0;
      i32x4 g3; g3[0] = 0; g3[1] = 0; g3[2] = 0; g3[3] = 0;
      i32x8 gz; gz[0] = 0; gz[1] = 0; gz[2] = 0; gz[3] = 0;
      gz[4] = 0; gz[5] = 0; gz[6] = 0; gz[7] = 0;
      u32x4 g0x = tdm_group0((uint32_t)(uintptr_t)xs, (uint64_t)(uintptr_t)(x + eoff));
      u32x4 g0y = tdm_group0((uint32_t)(uintptr_t)ys, (uint64_t)(uintptr_t)(y + eoff));
      __builtin_amdgcn_tensor_load_to_lds(g0x, g1, g2, g3, gz, 0);
      __builtin_amdgcn_tensor_load_to_lds(g0y, g1, g2, g3, gz, 0);
      __builtin_amdgcn_s_wait_tensorcnt(0);
    }
  } else {
    // Boundary tiles: bounds-checked fallback (zero pad outside the volume).
    for (int i = tid; i < HALO_N; i += 256) {
      const int wx = i % HW;
      const int t1 = i / HW;
      const int hy = t1 % HHy;
      const int dz = t1 / HHy;
      const int gd = d0 - 4 + dz, gh = h0 - 4 + hy, gw = w0 - 4 + wx;
      const bool ok = ((unsigned)gd < (unsigned)D_) &&
                      ((unsigned)gh < (unsigned)H_) &&
                      ((unsigned)gw < (unsigned)W_);
      const size_t off = (((size_t)b * D_ + (ok ? gd : 0)) * H_ + (ok ? gh : 0)) * (size_t)W_ + (ok ? gw : 0);
      xs[i] = ok ? x[off] : 0.0f;
      ys[i] = ok ? y[off] : 0.0f;
    }
  }
  __syncthreads();

  // Banded ones matrix B[24x16] for the 9-tap W sliding sum:
  // out[n] = sum_{j=n..n+8} in[j]   ->   B[j][n] = (n <= j <= n+8)
  // fp32 16x16x4 B layout: lanes 0-15 hold column n with K={0,1} in v0/v1,
  // lanes 16-31 hold K={2,3}.
  v2f bb[6];
#pragma unroll
  for (int kc = 0; kc < 6; kc++) {
    const int j0 = kc * 4 + 2 * half;
    bb[kc][0] = (j0 >= nn && j0 <= nn + 8) ? 1.0f : 0.0f;
    bb[kc][1] = (j0 + 1 >= nn && j0 + 1 <= nn + 8) ? 1.0f : 0.0f;
  }

  float accum[5][16];   // per-thread box-sums: [channel][d], this thread owns (h,w)

#pragma unroll
  for (int c = 0; c < 5; c++) {
    // ---- B1: W-axis 9-tap box sum via chained V_WMMA_F32_16X16X4_F32 ----
    // Out[16 rows x 16 w] = In[16 x 24] * Band[24 x 16], K tiled by 4 (6 WMMAs).
    for (int rt = wave; rt < ROWS / 16; rt += 8) {      // wave-uniform trip count
      const int row = rt * 16 + nn;                      // row = dz*24 + hy
      v8f acc = {0.f, 0.f, 0.f, 0.f, 0.f, 0.f, 0.f, 0.f};
#pragma unroll
      for (int kc = 0; kc < 6; kc++) {
        const int base = row * HW + kc * 4 + 2 * half;   // A layout: K={0,1}/{2,3}
        const float x0 = xs[base], x1 = xs[base + 1];
        const float y0 = ys[base], y1 = ys[base + 1];
        v2f a;
        if (c == 0)      { a[0] = x0;      a[1] = x1;      }
        else if (c == 1) { a[0] = y0;      a[1] = y1;      }
        else if (c == 2) { a[0] = x0 * x0; a[1] = x1 * x1; }
        else if (c == 3) { a[0] = y0 * y0; a[1] = y1 * y1; }
        else             { a[0] = x0 * y0; a[1] = x1 * y1; }
        acc = __builtin_amdgcn_wmma_f32_16x16x4_f32(
            false, a, false, bb[kc], (short)0, acc, false, false);
      }
      // C/D layout: vgpr r holds M=r (lanes 0-15) / M=r+8 (lanes 16-31), N=nn
#pragma unroll
      for (int r = 0; r < 8; r++) {
        const int rowOut = rt * 16 + r + 8 * half;
        b1[rowOut * TW + nn] = acc[r];
      }
    }
    __syncthreads();

    // ---- B2: H-axis 9-tap box sum (per (dz,w) column) ----
    for (int col = tid; col < HD * TW; col += 256) {     // 384 columns
      const int dz = col >> 4, w = col & 15;
      float v[HHy];
#pragma unroll
      for (int hy = 0; hy < HHy; hy++) v[hy] = b1[(dz * HHy + hy) * TW + w];
#pragma unroll
      for (int h = 0; h < THh; h++) {
        float s = 0.0f;
#pragma unroll
        for (int j = 0; j < 9; j++) s += v[h + j];
        b2[(dz * THh + h) * TW + w] = s;
      }
    }
    __syncthreads();

    // ---- B3: D-axis 9-tap box sum -> registers (thread owns one (h,w)) ----
    {
      const int w = tid & 15, h = (tid >> 4) & 15;
      float v[HD];
#pragma unroll
      for (int dz = 0; dz < HD; dz++) v[dz] = b2[(dz * THh + h) * TW + w];
#pragma unroll
      for (int d = 0; d < TD; d++) {
        float s = 0.0f;
#pragma unroll
        for (int j = 0; j < 9; j++) s += v[d + j];
        accum[c][d] = s;
      }
    }
    // no barrier needed here: next B1 writes b1; prior b1 readers already
    // passed the post-B2 barrier, and b2 is protected by the post-B1 barrier
    // of the next channel iteration.
  }

  // -------- Finalize: cc per voxel, block reduce, one atomic --------
  float part = 0.0f;
#pragma unroll
  for (int d = 0; d < TD; d++) {
    const float sx = accum[0][d], sy = accum[1][d];
    const float sxx = accum[2][d], syy = accum[3][d], sxy = accum[4][d];
    const float cross = sxy - sx * sy / 729.0f;
    const float vx = fmaxf(sxx - sx * sx / 729.0f, 1e-5f);
    const float vy = fmaxf(syy - sy * sy / 729.0f, 1e-5f);
    part += cross * cross / (vx * vy);
  }
  red[tid] = part;
  __syncthreads();
  for (int s = 128; s > 0; s >>= 1) {
    if (tid < s) red[tid] += red[tid + s];
    __syncthreads();
  }
  if (tid == 0) atomicAdd(ws, red[0]);
}

__global__ void lncc_zero(float* ws) { ws[0] = 0.0f; }

__global__ void lncc_fin(const float* __restrict__ ws, float* __restrict__ out) {
  out[0] = -(ws[0] / NVOXF);
}

extern "C" void kernel_launch(void* const* d_in, const int* in_sizes, int n_in,
                              void* d_out, int out_size, void* d_ws, size_t ws_size,
                              hipStream_t stream) {
  (void)in_sizes; (void)n_in; (void)out_size; (void)ws_size;
  const float* x = (const float*)d_in[0];
  const float* y = (const float*)d_in[1];
  float* out = (float*)d_out;
  float* ws  = (float*)d_ws;

  const size_t shmem = (size_t)LDS_FLOATS * sizeof(float);  // 173,056 B (<320KB/WGP)
  (void)hipFuncSetAttribute(reinterpret_cast<const void*>(lncc_main),
                            hipFuncAttributeMaxDynamicSharedMemorySize,
                            (int)shmem);

  lncc_zero<<<1, 1, 0, stream>>>(ws);
  dim3 grid(W_ / TW, H_ / THh, B_ * (D_ / TD));   // 10 x 12 x 20 = 2400 tiles
  lncc_main<<<grid, 256, shmem, stream>>>(x, y, ws);
  lncc_fin<<<1, 1, 0, stream>>>(ws, out);
}